// GCN_14740327760488
// MI455X (gfx1250) — compile-verified
//
#include <hip/hip_runtime.h>

typedef float v2f __attribute__((ext_vector_type(2)));
typedef float v8f __attribute__((ext_vector_type(8)));

// ---------------- degree / normalization ----------------

__global__ void k_init_deg(float* deg, int n) {
    int i = blockIdx.x * blockDim.x + threadIdx.x;
    if (i < n) deg[i] = 1.0f;  // self-loop contributes 1
}

__global__ void k_count_deg(const int* __restrict__ dst, long e, float* deg) {
    long i = (long)blockIdx.x * blockDim.x + threadIdx.x;
    if (i < e) unsafeAtomicAdd(&deg[dst[i]], 1.0f);
}

__global__ void k_rsqrt_inplace(float* deg, int n) {
    int i = blockIdx.x * blockDim.x + threadIdx.x;
    if (i < n) deg[i] = rsqrtf(deg[i]);
}

// ---------------- WMMA fp32 GEMM: Out[n,NOUT] = act(A[n,K]) @ W[K,NOUT] ----------------
// act = identity, or relu(a + bias[k]) when RELU_BIAS (fused layer-2 input epilogue).
// One wave -> 16 rows x NOUT cols.  W staged in LDS, K-pair interleaved:
//   lds[(k>>1)*PITCH + 2*c + (k&1)] = W[k][c]   -> b-fragment is one ds_load_b64
//   (pairs fuse into ds_load_2addr_b64).
// PITCH % 64 == 32 so the two lane-halves (adjacent K-pairs) hit disjoint bank halves.
// All B-fragments of a k-step are loaded into distinct registers BEFORE the WMMA burst
// so the LDS loads pipeline behind a single s_wait_dscnt.

template <int K, int NOUT, bool RELU_BIAS>
__global__ __launch_bounds__(256) void k_gemm_wmma(const float* __restrict__ A,
                                                   const float* __restrict__ W,
                                                   const float* __restrict__ bias,
                                                   float* __restrict__ Out,
                                                   int nRows) {
    constexpr int PITCH = 2 * NOUT + 32;   // dwords per K-pair row
    constexpr int HK    = K / 2;           // K per staging half
    constexpr int PAIRS = K / 4;           // K-pairs per half
    constexpr int CT    = NOUT / 16;       // 16-wide column tiles
    __shared__ __align__(16) float lw[PAIRS * PITCH];

    const int lane  = threadIdx.x & 31;
    const int wave  = threadIdx.x >> 5;            // 0..7
    const int laneM = lane & 15;                   // A row / D col within tile
    const int hi    = lane >> 4;                   // lane half selects K-pair

    const int rowBase = blockIdx.x * 128 + wave * 16 + laneM;
    const int rowC    = rowBase < nRows ? rowBase : 0;  // clamp; stores are guarded

    v8f acc[CT] = {};

    for (int kh = 0; kh < 2; ++kh) {
        __syncthreads();
        // stage half of W into LDS (coalesced global read)
        for (int idx = threadIdx.x; idx < HK * NOUT; idx += 256) {
            int lk = idx / NOUT;
            int c  = idx - lk * NOUT;
            lw[(lk >> 1) * PITCH + 2 * c + (lk & 1)] = W[(kh * HK + lk) * NOUT + c];
        }
        __syncthreads();

        for (int k4 = 0; k4 < HK; k4 += 4) {
            const int kk = kh * HK + k4 + 2 * hi;  // global k of a.x
            v2f a;
            a.x = A[(long)rowC * K + kk];
            a.y = A[(long)rowC * K + kk + 1];
            if (RELU_BIAS) {
                a.x = fmaxf(a.x + bias[kk], 0.0f);
                a.y = fmaxf(a.y + bias[kk + 1], 0.0f);
            }
            const float* lb = &lw[((k4 >> 1) + hi) * PITCH];

            // issue all LDS fragment loads first (distinct regs -> one dscnt wait)
            v2f bfrag[CT];
#pragma unroll
            for (int ct = 0; ct < CT; ++ct)
                bfrag[ct] = *(const v2f*)&lb[2 * (ct * 16 + laneM)];

            // WMMA burst: no D->A/B reuse, so no hazard NOPs between these
#pragma unroll
            for (int ct = 0; ct < CT; ++ct)
                acc[ct] = __builtin_amdgcn_wmma_f32_16x16x4_f32(
                    false, a, false, bfrag[ct], (short)0, acc[ct], false, false);
        }
    }

    // D layout: VGPR r -> M = r + 8*hi, N = laneM (per ISA C/D table)
#pragma unroll
    for (int ct = 0; ct < CT; ++ct) {
#pragma unroll
        for (int r = 0; r < 8; ++r) {
            int m = blockIdx.x * 128 + wave * 16 + 8 * hi + r;
            if (m < nRows) Out[(long)m * NOUT + ct * 16 + laneM] = acc[ct][r];
        }
    }
}

// ---------------- self-loop init: out[i][f] = h[i][f]*dinv[i]^2 (+ bias[f]) ----------------

__global__ void k_self_init(const float* __restrict__ h, const float* __restrict__ dinv,
                            const float* __restrict__ bias, float* __restrict__ out,
                            long total, int logF) {
    long gid = (long)blockIdx.x * blockDim.x + threadIdx.x;
    if (gid >= total) return;
    int  i = (int)(gid >> logF);
    int  f = (int)(gid & ((1 << logF) - 1));
    float d = dinv[i];
    float v = h[gid] * d * d;
    if (bias) v += bias[f];
    out[gid] = v;
}

// ---------------- edge scatter: out[dst] += h[src] * dinv[src]*dinv[dst] ----------------
// One thread = (edge, 4-float chunk); a wave gathers 512B of h[src] contiguously.

template <int LOGF>
__global__ void k_scatter(const int* __restrict__ src, const int* __restrict__ dst,
                          const float* __restrict__ dinv, const float* __restrict__ h,
                          float* __restrict__ out, long nE) {
    constexpr int CH = (1 << LOGF) / 4;  // float4 chunks per edge
    long gid = (long)blockIdx.x * blockDim.x + threadIdx.x;
    long e   = gid >> (LOGF - 2);
    if (e >= nE) return;
    int   c4 = (int)(gid & (CH - 1));
    int   s  = src[e], d = dst[e];
    float w  = dinv[s] * dinv[d];
    float4 v = ((const float4*)(h + ((long)s << LOGF)))[c4];
    float* o = out + ((long)d << LOGF) + 4 * c4;
    unsafeAtomicAdd(o + 0, v.x * w);
    unsafeAtomicAdd(o + 1, v.y * w);
    unsafeAtomicAdd(o + 2, v.z * w);
    unsafeAtomicAdd(o + 3, v.w * w);
}

// ---------------- launch ----------------

extern "C" void kernel_launch(void* const* d_in, const int* in_sizes, int n_in,
                              void* d_out, int out_size, void* d_ws, size_t ws_size,
                              hipStream_t stream) {
    const float* x   = (const float*)d_in[0];
    const int*   ei  = (const int*)d_in[1];
    const float* W1  = (const float*)d_in[2];
    const float* b1  = (const float*)d_in[3];
    const float* W2  = (const float*)d_in[4];
    const float* b2  = (const float*)d_in[5];
    float*       out = (float*)d_out;

    const int  n = in_sizes[0] / 128;   // 100000
    const long E = in_sizes[1] / 2;     // 1600000
    const int* srcIdx = ei;             // edge_index[0]
    const int* dstIdx = ei + E;         // edge_index[1]

    float* ws   = (float*)d_ws;
    float* dinv = ws;                         // n        (deg -> rsqrt in place)
    float* h1   = dinv + n;                   // n*128
    float* agg1 = h1 + (long)n * 128;         // n*128
    float* h2   = agg1 + (long)n * 128;       // n*64

    const int B = 256;
    auto blk = [](long work, int b) { return (int)((work + b - 1) / b); };

    // normalization coefficients
    k_init_deg<<<blk(n, B), B, 0, stream>>>(dinv, n);
    k_count_deg<<<blk(E, B), B, 0, stream>>>(dstIdx, E, dinv);
    k_rsqrt_inplace<<<blk(n, B), B, 0, stream>>>(dinv, n);

    // layer 1: h1 = x @ W1 ; agg1 = scatter(norm * h1) ; (bias+relu fused into layer-2 A load)
    k_gemm_wmma<128, 128, false><<<blk(n, 128), 256, 0, stream>>>(x, W1, nullptr, h1, n);
    k_self_init<<<blk((long)n * 128, B), B, 0, stream>>>(h1, dinv, nullptr, agg1, (long)n * 128, 7);
    k_scatter<7><<<blk(E * 32, B), B, 0, stream>>>(srcIdx, dstIdx, dinv, h1, agg1, E);

    // layer 2: h2 = relu(agg1 + b1) @ W2 ; d_out = scatter(norm * h2) + b2
    k_gemm_wmma<128, 64, true><<<blk(n, 128), 256, 0, stream>>>(agg1, W2, b1, h2, n);
    k_self_init<<<blk((long)n * 64, B), B, 0, stream>>>(h2, dinv, b2, out, (long)n * 64, 6);
    k_scatter<6><<<blk(E * 16, B), B, 0, stream>>>(srcIdx, dstIdx, dinv, h2, out, E);
}